// ReferenceRhythmEncoder_31705448579841
// MI455X (gfx1250) — compile-verified
//
#include <hip/hip_runtime.h>
#include <hip/hip_bf16.h>

#define T_LEN  8000
#define N_MEL  80
#define BATCH  64
#define BINS   24
#define SORT_N 8192
#define NT     256

typedef __attribute__((ext_vector_type(2))) float v2f;
typedef __attribute__((ext_vector_type(8))) float v8f;
typedef __attribute__((ext_vector_type(4))) int   v4i;

#if __has_builtin(__builtin_amdgcn_global_load_async_to_lds_b128) && \
    __has_builtin(__builtin_amdgcn_s_wait_asynccnt)
#define USE_ASYNC_LDS 1
// The clang builtin's pointer params live in LangAS::cuda_device /
// cuda_shared, which HIP source cannot name. Declare the LLVM intrinsic
// directly with numeric address spaces instead (opaque pointers: only the
// address space matters at IR level).
typedef __attribute__((address_space(1))) const v4i* as1_v4i_cptr;
typedef __attribute__((address_space(3))) v4i*       as3_v4i_ptr;
extern "C" __device__ void
llvm_amdgcn_global_load_async_to_lds_b128(as1_v4i_cptr gsrc, as3_v4i_ptr ldst,
                                          int ioffset, int cpol)
    __asm("llvm.amdgcn.global.load.async.to.lds.b128");
#else
#define USE_ASYNC_LDS 0
#endif

// ---------------------------------------------------------------------------
// Kernel 1: energy[b,t] = mean over MEL of mel[b,t,:]  via V_WMMA_F32_16X16X4_F32
// One wave = 16 timesteps. A (16x4 fp32) per ISA layout:
//   lanes 0-15 hold rows M=0..15; VGPR0 = K{0|2}, VGPR1 = K{1|3} (kp = lane>>4)
// B = constant 1/80 in every element -> D[m][n] = mean(row m) replicated over n.
// D layout: VGPR j: lanes 0-15 -> M=j, lanes 16-31 -> M=8+j.
// ---------------------------------------------------------------------------
__global__ __launch_bounds__(128) void energy_wmma_kernel(const float* __restrict__ mel,
                                                          float* __restrict__ energy) {
  const int lane = threadIdx.x & 31;
  const int wave = threadIdx.x >> 5;
  const int b    = blockIdx.y;
  const int t0   = (blockIdx.x * 4 + wave) * 16;
  const int row  = lane & 15;
  const int kp   = lane >> 4;

  v8f c = {};
  v2f bfrag;
  bfrag.x = 1.0f / N_MEL;
  bfrag.y = 1.0f / N_MEL;

#if USE_ASYNC_LDS
  // Stage each wave's contiguous 16x80 fp32 tile (5120 B) into LDS with the
  // async DMA path (ASYNCcnt), fully coalesced b128 transfers.
  __shared__ __align__(16) float tile[4][16 * N_MEL];
  {
    const float* gsrc = mel + (size_t)b * ((size_t)T_LEN * N_MEL) + (size_t)t0 * N_MEL;
    for (int i = 0; i < 10; ++i) {  // 10 * 32 lanes * 16B = 5120B
      const float* g = gsrc + i * 128 + lane * 4;
      float*       l = &tile[wave][i * 128 + lane * 4];
      // Flat shared address low 32 bits == LDS byte offset (aperture layout),
      // so the inttoptr truncation to AS(3) is exact.
      llvm_amdgcn_global_load_async_to_lds_b128(
          (as1_v4i_cptr)(uintptr_t)(const void*)g,
          (as3_v4i_ptr)(uintptr_t)(void*)l, 0, 0);
    }
    __builtin_amdgcn_s_wait_asynccnt(0);
  }
  const float* abase = &tile[wave][row * N_MEL + 2 * kp];
#else
  const float* abase = mel + (size_t)b * ((size_t)T_LEN * N_MEL) +
                       (size_t)(t0 + row) * N_MEL + 2 * kp;
#endif

  for (int k0 = 0; k0 < N_MEL; k0 += 4) {
    v2f a = *(const v2f*)(abase + k0);  // 8B-aligned by construction
    c = __builtin_amdgcn_wmma_f32_16x16x4_f32(false, a, false, bfrag, (short)0, c,
                                              false, false);
  }

  float* erow = energy + (size_t)b * T_LEN + t0;
  if (lane == 0) {
    for (int j = 0; j < 8; ++j) erow[j] = c[j];
  } else if (lane == 16) {
    for (int j = 0; j < 8; ++j) erow[8 + j] = c[j];
  }
}

// ---------------------------------------------------------------------------
// Block-wide helpers (NT = 256 threads)
// ---------------------------------------------------------------------------
__device__ __forceinline__ float block_sum(float v, float* s, int tid) {
  __syncthreads();
  s[tid] = v;
  __syncthreads();
  for (int off = NT / 2; off > 0; off >>= 1) {
    if (tid < off) s[tid] += s[tid + off];
    __syncthreads();
  }
  float r = s[0];
  __syncthreads();
  return r;
}

__device__ __forceinline__ void bitonic_sort_8192(float* s, int tid) {
  for (int k = 2; k <= SORT_N; k <<= 1) {
    for (int j = k >> 1; j > 0; j >>= 1) {
      __syncthreads();
      for (int i = tid; i < SORT_N; i += NT) {
        int ixj = i ^ j;
        if (ixj > i) {
          float a = s[i], b = s[ixj];
          bool up = ((i & k) == 0);
          if (up ? (a > b) : (a < b)) { s[i] = b; s[ixj] = a; }
        }
      }
    }
  }
  __syncthreads();
}

__device__ __forceinline__ float lds_quantile(const float* s, int n, float q) {
  float pos = q * (float)(n - 1);
  int   i0  = (int)floorf(pos);
  float fr  = pos - (float)i0;
  int   i1  = (i0 + 1 < n) ? i0 + 1 : n - 1;
  return s[i0] * (1.0f - fr) + s[i1] * fr;
}

// ---------------------------------------------------------------------------
// Kernel 2: one block per batch row; all per-row stats + 24-bin trace.
// ---------------------------------------------------------------------------
__global__ __launch_bounds__(NT) void rhythm_stats_kernel(const float* __restrict__ mel,
                                                          const float* __restrict__ energy_g,
                                                          float* __restrict__ out,
                                                          int use_energy) {
  const int b   = blockIdx.x;
  const int tid = threadIdx.x;

  extern __shared__ float smem[];
  float* s_e    = smem;                      // [T_LEN]
  float* s_prog = smem + T_LEN;              // [T_LEN]
  float* s_sort = smem + 2 * T_LEN;          // [SORT_N]
  float* s_red  = smem + 2 * T_LEN + SORT_N; // [NT]

  // --- energy -> LDS, mean / unbiased std (two-pass) ---
  float psum = 0.0f;
  for (int t = tid; t < T_LEN; t += NT) {
    float e;
    if (use_energy) {
      e = energy_g[(size_t)b * T_LEN + t];
    } else {
      const float* p = mel + (size_t)b * ((size_t)T_LEN * N_MEL) + (size_t)t * N_MEL;
      float s = 0.0f;
      for (int m = 0; m < N_MEL; ++m) s += p[m];
      e = s * (1.0f / N_MEL);
    }
    s_e[t] = e;
    psum += e;
  }
  float mean = block_sum(psum, s_red, tid) * (1.0f / T_LEN);
  float pss = 0.0f;
  for (int t = tid; t < T_LEN; t += NT) {
    float d = s_e[t] - mean;
    pss += d * d;
  }
  float sd = fmaxf(sqrtf(block_sum(pss, s_red, tid) / (float)(T_LEN - 1)), 1e-6f);

  auto dlt = [&](int t) -> float {
    return (t <= 0 || t >= T_LEN) ? 0.0f : fabsf(s_e[t] - s_e[t - 1]);
  };
  auto lr = [&](int t) -> float {  // sliding mean k=5, zero pad, count_include_pad
    float s = 0.0f;
    for (int u = t - 2; u <= t + 2; ++u) s += dlt(u);
    return s * (1.0f / 5.0f);
  };
  auto bsf = [&](int t) -> float {  // sliding mean k=7
    float s = 0.0f;
    for (int u = t - 3; u <= t + 3; ++u) s += dlt(u);
    return s * (1.0f / 7.0f);
  };

  // --- quantile(delta, 0.35): exact via bitonic sort ---
  for (int i = tid; i < SORT_N; i += NT) s_sort[i] = (i < T_LEN) ? dlt(i) : __builtin_inff();
  bitonic_sort_8192(s_sort, tid);
  float dthr = lds_quantile(s_sort, T_LEN, 0.35f);
  __syncthreads();

  // --- quantile(boundary_strength, 0.75) ---
  for (int i = tid; i < SORT_N; i += NT) s_sort[i] = (i < T_LEN) ? bsf(i) : __builtin_inff();
  bitonic_sort_8192(s_sort, tid);
  float bthr = lds_quantile(s_sort, T_LEN, 0.75f);
  __syncthreads();

  auto zf = [&](int t) -> float { return (s_e[t] - mean) / sd; };
  auto pausef = [&](int t) -> bool { return (zf(t) <= -0.5f) && (dlt(t) <= dthr); };

  // --- progress = cumsum(speech)/max(total,1): chunked block scan ---
  const int C = SORT_N / NT;  // 32
  const int t_begin = tid * C;
  float cnt = 0.0f;
  for (int u = 0; u < C; ++u) {
    int t = t_begin + u;
    if (t < T_LEN && !pausef(t)) cnt += 1.0f;
  }
  __syncthreads();
  s_red[tid] = cnt;
  __syncthreads();
  for (int off = 1; off < NT; off <<= 1) {  // Hillis-Steele inclusive scan
    float add = (tid >= off) ? s_red[tid - off] : 0.0f;
    float v   = s_red[tid];
    __syncthreads();
    s_red[tid] = v + add;
    __syncthreads();
  }
  float run = s_red[tid] - cnt;  // exclusive prefix
  for (int u = 0; u < C; ++u) {
    int t = t_begin + u;
    if (t < T_LEN) {
      if (!pausef(t)) run += 1.0f;
      s_prog[t] = run;
    }
  }
  __syncthreads();
  float invtot = 1.0f / fmaxf(s_prog[T_LEN - 1], 1.0f);
  __syncthreads();
  for (int t = tid; t < T_LEN; t += NT) s_prog[t] *= invtot;
  __syncthreads();

  // --- scalar stats ---
  float c_pause = 0.0f, c_pstart = 0.0f, c_sstart = 0.0f, c_bev = 0.0f, c_voiced = 0.0f;
  for (int t = tid; t < T_LEN; t += NT) {
    bool p   = pausef(t);
    bool pm1 = (t == 0) ? false : pausef(t - 1);
    if (p) c_pause += 1.0f;
    if (p && !pm1) c_pstart += 1.0f;                    // pause run starts
    bool sm1 = (t == 0) ? false : !pm1;
    if (!p && !sm1) c_sstart += 1.0f;                   // speech run starts
    if (bsf(t) >= bthr) c_bev += 1.0f;
    if (zf(t) > -0.1f) c_voiced += 1.0f;
  }
  c_pause  = block_sum(c_pause, s_red, tid);
  c_pstart = block_sum(c_pstart, s_red, tid);
  c_sstart = block_sum(c_sstart, s_red, tid);
  c_bev    = block_sum(c_bev, s_red, tid);
  c_voiced = block_sum(c_voiced, s_red, tid);

  if (tid == 0) {
    float* st = out + (size_t)b * 6;
    float c_speech = (float)T_LEN - c_pause;
    st[0] = c_pause * (1.0f / T_LEN);
    st[1] = (c_pstart > 0.0f) ? c_pause / fmaxf(c_pstart, 1.0f) : 0.0f;
    st[2] = (c_sstart > 0.0f) ? c_speech / fmaxf(c_sstart, 1.0f) : 0.0f;
    st[3] = lr(T_LEN - 1) - lr(0);
    st[4] = c_bev * (1.0f / T_LEN);
    st[5] = c_voiced * (1.0f / T_LEN);
  }

  // --- 24-bin progress resampling (no __syncthreads below this point) ---
  if (tid < BINS) {
    float tgt = (float)tid * (1.0f / (BINS - 1));
    int lo = 0, hi = T_LEN;  // searchsorted(progress, tgt, side='left')
    while (lo < hi) {
      int mid = (lo + hi) >> 1;
      if (s_prog[mid] < tgt) lo = mid + 1; else hi = mid;
    }
    int right = lo;
    int left  = right - 1;
    if (left < 0) left = 0;
    if (left > T_LEN - 1) left = T_LEN - 1;
    int rc = (right < T_LEN - 1) ? right : T_LEN - 1;
    if (rc < 0) rc = 0;
    float pl = s_prog[left], pr = s_prog[rc];
    float denom = fmaxf(fabsf(pr - pl), 1e-6f);
    float alpha = fminf(fmaxf((tgt - pl) / denom, 0.0f), 1.0f);
    if (right >= T_LEN) alpha = 1.0f;
    if (right <= 0) alpha = 0.0f;

    auto feat = [&](int t, int i) -> float {
      switch (i) {
        case 0: return pausef(t) ? 1.0f : 0.0f;
        case 1: return lr(t);
        case 2: return (bsf(t) >= bthr) ? 1.0f : 0.0f;
        case 3: return s_prog[t] - (float)t * (1.0f / (T_LEN - 1));
        default: return (zf(t) > -0.1f) ? 1.0f : 0.0f;
      }
    };
    float* tr = out + (size_t)BATCH * 6 + ((size_t)b * BINS + tid) * 5;
    for (int i = 0; i < 5; ++i)
      tr[i] = feat(left, i) * (1.0f - alpha) + feat(rc, i) * alpha;
  }
}

// ---------------------------------------------------------------------------
extern "C" void kernel_launch(void* const* d_in, const int* in_sizes, int n_in,
                              void* d_out, int out_size, void* d_ws, size_t ws_size,
                              hipStream_t stream) {
  const float* mel = (const float*)d_in[0];
  float* out = (float*)d_out;

  const size_t energy_bytes = (size_t)BATCH * T_LEN * sizeof(float);
  int use_energy = (d_ws != nullptr && ws_size >= energy_bytes) ? 1 : 0;
  float* energy = (float*)d_ws;

  if (use_energy) {
    dim3 grid(T_LEN / 64, BATCH);  // 128 threads = 4 waves, 16 timesteps/wave
    energy_wmma_kernel<<<grid, dim3(128), 0, stream>>>(mel, energy);
  }

  size_t shmem = (size_t)(2 * T_LEN + SORT_N + NT) * sizeof(float);  // ~97.8 KB
  rhythm_stats_kernel<<<dim3(BATCH), dim3(NT), shmem, stream>>>(mel, energy, out,
                                                                use_energy);
}